// Criterion_37984690765901
// MI455X (gfx1250) — compile-verified
//
#include <hip/hip_runtime.h>
#include <hip/hip_bf16.h>

// Problem constants (match reference)
constexpr int Bn = 65536;   // batch
constexpr int Dn = 128;     // feature dim
constexpr int Cn = 10;      // classes

typedef float v2f __attribute__((ext_vector_type(2)));
typedef float v8f __attribute__((ext_vector_type(8)));

// ---------------- wave32 butterfly reductions ----------------
__device__ __forceinline__ float wave_sum32(float v) {
#pragma unroll
    for (int off = 16; off > 0; off >>= 1) v += __shfl_xor(v, off, 32);
    return v;
}

// Per-row KL contribution: sum_d t_d*(log t_d - log_p_d)
//   t = softmax(enc), log_p = log_softmax(prior)
// = S1/Se - max_e - log Se + max_p + log Sp,  S1 = sum e_d*(enc_d - prior_d)
__device__ __forceinline__ float row_kld(const float enc[4], const float pri[4]) {
    float me = fmaxf(fmaxf(enc[0], enc[1]), fmaxf(enc[2], enc[3]));
    float mp = fmaxf(fmaxf(pri[0], pri[1]), fmaxf(pri[2], pri[3]));
#pragma unroll
    for (int off = 16; off > 0; off >>= 1) {
        me = fmaxf(me, __shfl_xor(me, off, 32));
        mp = fmaxf(mp, __shfl_xor(mp, off, 32));
    }
    float se = 0.f, sp = 0.f, s1 = 0.f;
#pragma unroll
    for (int j = 0; j < 4; ++j) {
        float ee = __expf(enc[j] - me);
        se += ee;
        s1 += ee * (enc[j] - pri[j]);
        sp += __expf(pri[j] - mp);
    }
#pragma unroll
    for (int off = 16; off > 0; off >>= 1) {
        se += __shfl_xor(se, off, 32);
        sp += __shfl_xor(sp, off, 32);
        s1 += __shfl_xor(s1, off, 32);
    }
    return s1 / se - me - __logf(se) + mp + __logf(sp);
}

// ---------------- main KLD streaming kernel ----------------
// One wave per row; lane holds float4 -> full 128-elem row in the wave.
__global__ void kld_kernel(const float* __restrict__ mean_t, const float* __restrict__ mean_s,
                           const float* __restrict__ lstd_t, const float* __restrict__ lstd_s,
                           const float* __restrict__ epr_t,  const float* __restrict__ epr_s,
                           const float* __restrict__ eps_t,  const float* __restrict__ eps_s,
                           double* __restrict__ acc) {
    const int lane   = threadIdx.x & 31;
    const int wave   = (blockIdx.x * blockDim.x + threadIdx.x) >> 5;
    const int nwaves = (gridDim.x * blockDim.x) >> 5;

    float acc_t = 0.f, acc_s = 0.f;
    for (int r = wave; r < Bn; r += nwaves) {
        const int i4 = r * (Dn / 4) + lane;  // float4 index into row
        // ---- t branch: prior = eps_prior_t (mean 0) ----
        {
            float4 m = ((const float4*)mean_t)[i4];
            float4 l = ((const float4*)lstd_t)[i4];
            float4 e = ((const float4*)eps_t)[i4];
            float4 p = ((const float4*)epr_t)[i4];
            float enc[4] = { m.x + __expf(0.5f * l.x) * e.x,
                             m.y + __expf(0.5f * l.y) * e.y,
                             m.z + __expf(0.5f * l.z) * e.z,
                             m.w + __expf(0.5f * l.w) * e.w };
            float pri[4] = { p.x, p.y, p.z, p.w };
            acc_t += row_kld(enc, pri);
        }
        // ---- s branch: prior = 1 + eps_prior_s ----
        {
            float4 m = ((const float4*)mean_s)[i4];
            float4 l = ((const float4*)lstd_s)[i4];
            float4 e = ((const float4*)eps_s)[i4];
            float4 p = ((const float4*)epr_s)[i4];
            float enc[4] = { m.x + __expf(0.5f * l.x) * e.x,
                             m.y + __expf(0.5f * l.y) * e.y,
                             m.z + __expf(0.5f * l.z) * e.z,
                             m.w + __expf(0.5f * l.w) * e.w };
            float pri[4] = { 1.f + p.x, 1.f + p.y, 1.f + p.z, 1.f + p.w };
            acc_s += row_kld(enc, pri);
        }
    }
    // values are wave-uniform after the butterflies; lane 0 commits
    if (lane == 0) {
        atomicAdd(&acc[2], (double)acc_t);
        atomicAdd(&acc[3], (double)acc_s);
    }
}

// ---------------- cross-entropy + entropy kernel ----------------
__global__ void ce_ent_kernel(const float* __restrict__ y, const float* __restrict__ s,
                              const int* __restrict__ tgt, double* __restrict__ acc) {
    const int r = blockIdx.x * blockDim.x + threadIdx.x;
    float lt = 0.f, ent = 0.f;
    if (r < Bn) {
        const float* yr = y + r * Cn;
        float v[Cn];
#pragma unroll
        for (int c = 0; c < Cn; ++c) v[c] = yr[c];
        float m = v[0];
#pragma unroll
        for (int c = 1; c < Cn; ++c) m = fmaxf(m, v[c]);
        float se = 0.f;
#pragma unroll
        for (int c = 0; c < Cn; ++c) se += __expf(v[c] - m);
        const float lse = m + __logf(se);
        const int t = tgt[r];
        float ysel = 0.f;
#pragma unroll
        for (int c = 0; c < Cn; ++c) ysel = (c == t) ? v[c] : ysel;
        lt = lse - ysel;  // -log softmax(y)[target]

        float2 ab = ((const float2*)s)[r];
        float mm = fmaxf(ab.x, ab.y);
        float ea = __expf(ab.x - mm), eb = __expf(ab.y - mm);
        float Z = ea + eb;
        float lz = mm + __logf(Z);
        float pa = ea / Z, pb = eb / Z;
        ent = -(pa * (ab.x - lz) + pb * (ab.y - lz));  // -sum p*log p
    }
    lt  = wave_sum32(lt);
    ent = wave_sum32(ent);
    if ((threadIdx.x & 31) == 0) {
        atomicAdd(&acc[0], (double)lt);
        atomicAdd(&acc[1], (double)ent);
    }
}

// ---------------- finalize: weighted combine via WMMA ----------------
// out = L_t + lam_e*Loss_e + lam_od*(L_zt + L_zs), lam = c * 2^(step/1000)
// The 4-term weighted sum is computed as D = A x ones with A row 0 holding the
// pre-scaled terms (ISA 16x4 f32 A layout: lane0 -> K=0,1 ; lane16 -> K=2,3).
__global__ void finalize_kernel(const double* __restrict__ acc, const int* __restrict__ step_p,
                                float* __restrict__ out) {
    const int lane = threadIdx.x;   // 32 threads, EXEC all ones, branch-free pre-WMMA
    const double invB = 1.0 / (double)Bn;
    const float frac = (float)step_p[0] * (1.0f / 1000.0f);
    const float g = exp2f(frac);          // GAMMA=2 -> gamma^frac = 2^frac
    const float lam_e = 0.1f * g, lam_od = 0.036f * g;

    const float t0 = (float)(acc[0] * invB);            // L_t
    const float t1 = (float)(acc[1] * invB) * lam_e;    // lam_e * Loss_e
    const float t2 = (float)(acc[2] * invB) * lam_od;   // lam_od * L_zt
    const float t3 = (float)(acc[3] * invB) * lam_od;   // lam_od * L_zs

#if __has_builtin(__builtin_amdgcn_wmma_f32_16x16x4_f32)
    v2f A;
    A.x = (lane == 0) ? t0 : ((lane == 16) ? t2 : 0.f);
    A.y = (lane == 0) ? t1 : ((lane == 16) ? t3 : 0.f);
    v2f Bv;  Bv.x = 1.f;  Bv.y = 1.f;   // all-ones B: layout-robust
    v8f C = {};
    C = __builtin_amdgcn_wmma_f32_16x16x4_f32(
            /*neg_a=*/false, A, /*neg_b=*/false, Bv,
            /*c_mod=*/(short)0, C, /*reuse_a=*/false, /*reuse_b=*/false);
    if (lane == 0) out[0] = C[0];       // D[0][0] = t0+t1+t2+t3
#else
    if (lane == 0) out[0] = t0 + t1 + t2 + t3;
#endif
}

// ---------------- launcher ----------------
extern "C" void kernel_launch(void* const* d_in, const int* in_sizes, int n_in,
                              void* d_out, int out_size, void* d_ws, size_t ws_size,
                              hipStream_t stream) {
    const float* mean_t      = (const float*)d_in[0];
    const float* mean_s      = (const float*)d_in[1];
    const float* log_std_t   = (const float*)d_in[2];
    const float* log_std_s   = (const float*)d_in[3];
    const float* y_zt        = (const float*)d_in[4];
    const float* s_zt        = (const float*)d_in[5];
    const float* eps_prior_t = (const float*)d_in[6];
    const float* eps_prior_s = (const float*)d_in[7];
    const float* eps_t       = (const float*)d_in[8];
    const float* eps_s       = (const float*)d_in[9];
    const int*   target      = (const int*)d_in[10];
    const int*   step        = (const int*)d_in[11];
    double* acc = (double*)d_ws;   // [Lt_sum, ent_sum, zt_sum, zs_sum]

    hipMemsetAsync(acc, 0, 4 * sizeof(double), stream);

    kld_kernel<<<2048, 256, 0, stream>>>(mean_t, mean_s, log_std_t, log_std_s,
                                         eps_prior_t, eps_prior_s, eps_t, eps_s, acc);
    ce_ent_kernel<<<Bn / 256, 256, 0, stream>>>(y_zt, s_zt, target, acc);
    finalize_kernel<<<1, 32, 0, stream>>>(acc, step, (float*)d_out);
}